// DLEM_54443005444584
// MI455X (gfx1250) — compile-verified
//
#include <hip/hip_runtime.h>
#include <hip/hip_bf16.h>

// ---------------------------------------------------------------------------
// CDNA5 (gfx1250) implementation.
// All convolutions (conv1d and conv_transpose1d) are implicit GEMMs executed
// with V_WMMA_F32_16X16X4_F32 (fp32-precise matrix core path, wave32).
// Each wave computes a 2x2 grid of 16x16 tiles (32 co x 32 n) so every fetched
// A/B operand feeds two WMMAs -> half the operand traffic per matrix op, and
// four independent accumulator chains for the scheduler to interleave.
// The 510-step dlem scan runs one workgroup per batch row entirely in LDS.
// ---------------------------------------------------------------------------

typedef float v2f __attribute__((ext_vector_type(2)));
typedef float v8f __attribute__((ext_vector_type(8)));

#define TILE_N    16
#define WAVES_PB  4
#define WAVE_N    32                    // columns per wave (2 N-tiles)
#define BLK_N     (WAVE_N * WAVES_PB)   // 128 output columns per block
#define BLK_M     32                    // output channels per block (2 M-tiles)
#define XS_STRIDE 132                   // staged row stride (130 used + pad)

#define NB   64
#define NN   2048

// ---------------------------------------------------------------------------
// Generic 1-D conv / transpose-conv as WMMA GEMM.
//   Y[b, co, n] = relu(bias[co] + sum_{ci,k} W[co*wco + ci*wci + k*wk] *
//                                  X[b, ci, n + k + col_off])   (X OOB -> 0)
// A-matrix 16x4 layout: lanes 0-15 -> M=lane, K={0,1}; lanes 16-31 -> K={2,3}.
// B-matrix 4x16 mirrors it (N per lane). C/D: VGPR r -> M=r (+8 for hi lanes).
// ---------------------------------------------------------------------------
__global__ void conv1d_wmma(const float* __restrict__ X, long xbs, int Lin, int Cin,
                            const float* __restrict__ W, int wco, int wci, int wk,
                            const float* __restrict__ bias,
                            float* __restrict__ Y, long ybs, int Lout, int Cout,
                            int col_off)
{
    extern __shared__ float xs[];                 // Cin * XS_STRIDE floats
    const int b       = blockIdx.z;
    const int co_base = blockIdx.y * BLK_M;
    const int p0      = blockIdx.x * BLK_N;
    const int tid     = threadIdx.x;
    const int wave    = tid >> 5;
    const int lane    = tid & 31;

    const float* Xb = X + (long)b * xbs;

    // Stage input tile: columns [p0+col_off, p0+col_off+130), zero-padded OOB.
    const int ncols  = BLK_N + 2;
    const int nstage = Cin * ncols;
    for (int idx = tid; idx < nstage; idx += blockDim.x) {
        int ci = idx / ncols;
        int c  = idx - ci * ncols;
        int gc = p0 + col_off + c;
        float v = 0.0f;
        if (gc >= 0 && gc < Lin) v = Xb[(long)ci * Lin + gc];
        xs[ci * XS_STRIDE + c] = v;
    }
    __syncthreads();

    v8f acc00 = {}, acc01 = {}, acc10 = {}, acc11 = {};
    const int m     = lane & 15;           // M row (A) / N col (B,C)
    const int kk    = (lane >> 4) * 2;     // K sub-offset within the 4-step
    const int colb0 = wave * WAVE_N + m;   // B column, N-tile 0
    const int colb1 = colb0 + TILE_N;      // B column, N-tile 1
    const int arow0 = (co_base + m) * wco;          // A row, M-tile 0
    const int arow1 = (co_base + TILE_N + m) * wco; // A row, M-tile 1

    #pragma unroll
    for (int k = 0; k < 3; ++k) {
        for (int ci0 = 0; ci0 < Cin; ci0 += 4) {
            const int ci = ci0 + kk;
            v2f a0, a1, b0, b1;
            const int widx = ci * wci + k * wk;
            a0.x = W[arow0 + widx];
            a0.y = W[arow0 + widx + wci];
            a1.x = W[arow1 + widx];
            a1.y = W[arow1 + widx + wci];
            b0.x = xs[(ci    ) * XS_STRIDE + colb0 + k];
            b0.y = xs[(ci + 1) * XS_STRIDE + colb0 + k];
            b1.x = xs[(ci    ) * XS_STRIDE + colb1 + k];
            b1.y = xs[(ci + 1) * XS_STRIDE + colb1 + k];
            acc00 = __builtin_amdgcn_wmma_f32_16x16x4_f32(false, a0, false, b0, (short)0, acc00, false, false);
            acc01 = __builtin_amdgcn_wmma_f32_16x16x4_f32(false, a0, false, b1, (short)0, acc01, false, false);
            acc10 = __builtin_amdgcn_wmma_f32_16x16x4_f32(false, a1, false, b0, (short)0, acc10, false, false);
            acc11 = __builtin_amdgcn_wmma_f32_16x16x4_f32(false, a1, false, b1, (short)0, acc11, false, false);
        }
    }

    // Bias + ReLU + store. VGPR r holds M = r (lanes 0-15) / r+8 (lanes 16-31).
    float* Yb = Y + (long)b * ybs;
    const int mhi = (lane < 16) ? 0 : 8;
    const int n0  = p0 + wave * WAVE_N + m;
    const int n1  = n0 + TILE_N;
    #pragma unroll
    for (int r = 0; r < 8; ++r) {
        const int co0 = co_base + mhi + r;
        const int co1 = co0 + TILE_N;
        const float bia0 = bias[co0];
        const float bia1 = bias[co1];
        if (n0 < Lout) {
            float v00 = acc00[r] + bia0;
            float v10 = acc10[r] + bia1;
            Yb[(long)co0 * Lout + n0] = v00 > 0.0f ? v00 : 0.0f;
            Yb[(long)co1 * Lout + n0] = v10 > 0.0f ? v10 : 0.0f;
        }
        if (n1 < Lout) {
            float v01 = acc01[r] + bia0;
            float v11 = acc11[r] + bia1;
            Yb[(long)co0 * Lout + n1] = v01 > 0.0f ? v01 : 0.0f;
            Yb[(long)co1 * Lout + n1] = v11 > 0.0f ? v11 : 0.0f;
        }
    }
}

// ---------------------------------------------------------------------------
// Concatenate signal (128 ch) + seq (32 ch) into X0 (64,160,2048).
// ---------------------------------------------------------------------------
__global__ void concat_kernel(const float* __restrict__ sig,
                              const float* __restrict__ seq,
                              float* __restrict__ X0)
{
    long idx = (long)blockIdx.x * blockDim.x + threadIdx.x;
    const long total = (long)NB * 160 * NN;
    if (idx >= total) return;
    long n  = idx & (NN - 1);
    long t  = idx >> 11;
    long ci = t % 160;
    long b  = t / 160;
    float v = (ci < 128) ? sig[(b * 128 + ci) * NN + n]
                         : seq[(b * 32 + (ci - 128)) * NN + n];
    X0[idx] = v;
}

// ---------------------------------------------------------------------------
// 1x1 mixer conv (128 -> 2) + sigmoid. LOg: (64, 32, 2048) each.
// ---------------------------------------------------------------------------
__global__ void mixer_kernel(const float* __restrict__ LO0, const float* __restrict__ LO1,
                             const float* __restrict__ LO2, const float* __restrict__ LO3,
                             const float* __restrict__ mw, const float* __restrict__ mb,
                             float* __restrict__ Lb, float* __restrict__ Rb)
{
    int idx = blockIdx.x * blockDim.x + threadIdx.x;
    if (idx >= NB * NN) return;
    int b = idx >> 11;
    int n = idx & (NN - 1);
    const float* los[4] = {LO0, LO1, LO2, LO3};
    float z0 = mb[0], z1 = mb[1];
    #pragma unroll
    for (int g = 0; g < 4; ++g) {
        const float* p = los[g] + (long)b * 32 * NN + n;
        for (int c = 0; c < 32; ++c) {
            float v = p[(long)c * NN];
            z0 += mw[g * 32 + c] * v;
            z1 += mw[128 + g * 32 + c] * v;
        }
    }
    Lb[idx] = 1.0f / (1.0f + __expf(-z0));
    Rb[idx] = 1.0f / (1.0f + __expf(-z1));
}

// ---------------------------------------------------------------------------
// dlem scan: one workgroup per batch row; 510 barrier-separated steps in LDS.
// ---------------------------------------------------------------------------
__global__ void dlem_scan_kernel(const float* __restrict__ curr_diag,
                                 const float* __restrict__ Lb,
                                 const float* __restrict__ Rb,
                                 const float* __restrict__ cptr,
                                 float* __restrict__ out)
{
    __shared__ float Ls[NN];
    __shared__ float Rs[NN + 1];
    __shared__ float bufA[NN + 1];
    __shared__ float bufB[NN + 1];
    const int b   = blockIdx.x;
    const int tid = threadIdx.x;
    const float konst = cptr[0];

    for (int n = tid; n < NN; n += blockDim.x) {
        Ls[n]   = Lb[b * NN + n];
        Rs[n]   = Rb[b * NN + n];
        bufA[n] = (n < NN - 1) ? curr_diag[b * (NN - 1) + n] : 0.0f;
    }
    if (tid == 0) { Rs[NN] = 0.0f; bufA[NN] = 0.0f; bufB[NN] = 0.0f; }
    __syncthreads();

    float* cur = bufA;
    float* nxt = bufB;
    for (int d = 1; d <= 510; ++d) {
        for (int n = tid; n < NN; n += blockDim.x) {
            float a1  = cur[n + 1];
            float r1  = Rs[n + 1];
            int   i0  = d + n;
            float ld  = (i0     < NN) ? Ls[i0]     : 0.0f;
            float ld1 = (i0 + 1 < NN) ? Ls[i0 + 1] : 0.0f;
            nxt[n] = konst * (a1 * r1 + cur[n] * ld) / (Rs[n] + ld1);
        }
        __syncthreads();
        float* t = cur; cur = nxt; nxt = t;
    }
    for (int n = tid; n < NN - 511; n += blockDim.x)
        out[b * (NN - 511) + n] = cur[n];
}

// ---------------------------------------------------------------------------
// Host-side launch
// ---------------------------------------------------------------------------
static inline void run_conv(hipStream_t s,
                            const float* X, long xbs, int Lin, int Cin,
                            const float* W, int wco, int wci, int wk,
                            const float* bias,
                            float* Y, long ybs, int Lout, int Cout, int col_off)
{
    dim3 grid((Lout + BLK_N - 1) / BLK_N, Cout / BLK_M, NB);
    size_t sh = (size_t)Cin * XS_STRIDE * sizeof(float);
    conv1d_wmma<<<grid, dim3(32 * WAVES_PB), sh, s>>>(X, xbs, Lin, Cin, W, wco, wci, wk,
                                                      bias, Y, ybs, Lout, Cout, col_off);
}

// transpose-conv: A[co,ci,k] = w[ci, co, 2-k] -> base w+2, strides (3, 96, -1)
static inline void run_tconv(hipStream_t s,
                             const float* X, long xbs, int Lin,
                             const float* tw, const float* tb,
                             float* Y)
{
    const int Lout = Lin + 2;
    run_conv(s, X, xbs, Lin, 32, tw + 2, 3, 96, -1, tb, Y, (long)32 * Lout, Lout, 32, -2);
}

extern "C" void kernel_launch(void* const* d_in, const int* in_sizes, int n_in,
                              void* d_out, int out_size, void* d_ws, size_t ws_size,
                              hipStream_t stream)
{
    const float* signal    = (const float*)d_in[0];
    const float* seq       = (const float*)d_in[1];
    const float* curr_diag = (const float*)d_in[2];
    const float* kconst    = (const float*)d_in[3];
    const float* c0w = (const float*)d_in[4];  const float* c0b = (const float*)d_in[5];
    const float* c1w = (const float*)d_in[6];  const float* c1b = (const float*)d_in[7];
    const float* c2w = (const float*)d_in[8];  const float* c2b = (const float*)d_in[9];
    const float* c3w = (const float*)d_in[10]; const float* c3b = (const float*)d_in[11];
    const float* t0w = (const float*)d_in[12]; const float* t0b = (const float*)d_in[13];
    const float* t1w = (const float*)d_in[14]; const float* t1b = (const float*)d_in[15];
    const float* t2w = (const float*)d_in[16]; const float* t2b = (const float*)d_in[17];
    const float* t3w = (const float*)d_in[18]; const float* t3b = (const float*)d_in[19];
    const float* mw  = (const float*)d_in[20]; const float* mb  = (const float*)d_in[21];
    float* out = (float*)d_out;

    // ---- workspace carve-up (floats) ----
    float* ws = (float*)d_ws;
    size_t off = 0;
    float* X0  = ws + off; off += (size_t)NB * 160 * 2048;
    float* T0  = ws + off; off += (size_t)NB * 128 * 2046;
    float* T1  = ws + off; off += (size_t)NB * 96  * 2044;
    float* T2  = ws + off; off += (size_t)NB * 64  * 2042;
    float* T3  = ws + off; off += (size_t)NB * 32  * 2040;
    float* LO0 = ws + off; off += (size_t)NB * 32 * 2048;
    float* LO1 = ws + off; off += (size_t)NB * 32 * 2048;
    float* LO2 = ws + off; off += (size_t)NB * 32 * 2048;
    float* LO3 = ws + off; off += (size_t)NB * 32 * 2048;
    float* TA  = ws + off; off += (size_t)NB * 32 * 2048;
    float* TB  = ws + off; off += (size_t)NB * 32 * 2048;
    float* Lb  = ws + off; off += (size_t)NB * 2048;
    float* Rb  = ws + off; off += (size_t)NB * 2048;

    // ---- 1) concat signal+seq ----
    {
        long total = (long)NB * 160 * 2048;
        concat_kernel<<<dim3((unsigned)((total + 255) / 256)), dim3(256), 0, stream>>>(signal, seq, X0);
    }

    // ---- 2) conv stack (WMMA GEMMs) ----
    run_conv(stream, X0,            (long)160 * 2048, 2048, 160, c0w, 160 * 3, 3, 1, c0b, T0, (long)128 * 2046, 2046, 128, 0);
    run_conv(stream, T0 + 32 * 2046,(long)128 * 2046, 2046,  96, c1w,  96 * 3, 3, 1, c1b, T1, (long)96  * 2044, 2044,  96, 0);
    run_conv(stream, T1 + 32 * 2044,(long)96  * 2044, 2044,  64, c2w,  64 * 3, 3, 1, c2b, T2, (long)64  * 2042, 2042,  64, 0);
    run_conv(stream, T2 + 32 * 2042,(long)64  * 2042, 2042,  32, c3w,  32 * 3, 3, 1, c3b, T3, (long)32  * 2040, 2040,  32, 0);

    // ---- 3) transpose-conv chains back to N=2048 ----
    // layer 0: tconv3
    run_tconv(stream, T0, (long)128 * 2046, 2046, t3w, t3b, LO0);
    // layer 1: tconv2 -> tconv3
    run_tconv(stream, T1, (long)96 * 2044, 2044, t2w, t2b, TA);
    run_tconv(stream, TA, (long)32 * 2046, 2046, t3w, t3b, LO1);
    // layer 2: tconv1 -> tconv2 -> tconv3
    run_tconv(stream, T2, (long)64 * 2042, 2042, t1w, t1b, TA);
    run_tconv(stream, TA, (long)32 * 2044, 2044, t2w, t2b, TB);
    run_tconv(stream, TB, (long)32 * 2046, 2046, t3w, t3b, LO2);
    // layer 3: tconv0 -> tconv1 -> tconv2 -> tconv3
    run_tconv(stream, T3, (long)32 * 2040, 2040, t0w, t0b, TA);
    run_tconv(stream, TA, (long)32 * 2042, 2042, t1w, t1b, TB);
    run_tconv(stream, TB, (long)32 * 2044, 2044, t2w, t2b, TA);
    run_tconv(stream, TA, (long)32 * 2046, 2046, t3w, t3b, LO3);

    // ---- 4) mixer + sigmoid ----
    mixer_kernel<<<dim3((NB * NN + 255) / 256), dim3(256), 0, stream>>>(
        LO0, LO1, LO2, LO3, mw, mb, Lb, Rb);

    // ---- 5) dlem scan: one WG per batch row, 510 LDS-resident steps ----
    dlem_scan_kernel<<<dim3(NB), dim3(256), 0, stream>>>(curr_diag, Lb, Rb, kconst, out);

    (void)in_sizes; (void)n_in; (void)out_size; (void)ws_size;
}